// SelfAttention_23570780521247
// MI455X (gfx1250) — compile-verified
//
#include <hip/hip_runtime.h>
#include <math.h>

typedef float v2f __attribute__((ext_vector_type(2)));
typedef float v8f __attribute__((ext_vector_type(8)));
typedef int   v4i __attribute__((ext_vector_type(4)));

#define B_     32
#define DIM_   4096
#define NH_    32
#define NKV_   8
#define HD_    128
#define L_     4096   // start_pos + 1
#define NREP_  4      // NH / NKV
#define SPLIT_ 4      // flash-decode key-range split
#define TILES_ (L_ / SPLIT_ / 16)
#define KVSTRIDE_ (NKV_ * HD_)   // floats between consecutive cache rows

#define AS1 __attribute__((address_space(1)))
#define AS3 __attribute__((address_space(3)))

#if __has_builtin(__builtin_amdgcn_global_load_async_to_lds_b128)
#define HAVE_ASYNC_LDS 1
#else
#define HAVE_ASYNC_LDS 0
#endif

// Copy 16B/lane from global to LDS via the CDNA5 async path (ASYNCcnt).
__device__ __forceinline__ void async_b128(const float* g, float* l) {
#if HAVE_ASYNC_LDS
  __builtin_amdgcn_global_load_async_to_lds_b128((AS1 v4i*)g, (AS3 v4i*)l, 0, 0);
#else
  *(float4*)l = *(const float4*)g;   // sync fallback (DS in-order per wave)
#endif
}

template <int N> __device__ __forceinline__ void wait_async() {
#if __has_builtin(__builtin_amdgcn_s_wait_asynccnt)
  __builtin_amdgcn_s_wait_asynccnt(N);
#elif HAVE_ASYNC_LDS
  asm volatile("s_wait_asynccnt %0" :: "i"(N) : "memory");
#endif
}

__device__ __forceinline__ void wait_dscnt0() {
#if __has_builtin(__builtin_amdgcn_s_wait_dscnt)
  __builtin_amdgcn_s_wait_dscnt(0);
#else
  asm volatile("s_wait_dscnt 0" ::: "memory");
#endif
}

// ---------------------------------------------------------------------------
// Y(32 x N) = X(32 x K) * W(N x K)^T   via V_WMMA_F32_16X16X4_F32
// ---------------------------------------------------------------------------
__global__ __launch_bounds__(32) void gemm_xwt(const float* __restrict__ X,
                                               const float* __restrict__ W,
                                               float* __restrict__ Y,
                                               int K, int N) {
  const int nt   = blockIdx.x;
  const int lane = threadIdx.x;
  const int m    = lane & 15;
  const int hi   = lane >> 4;

  const float* xr0 = X + (size_t)m * K;
  const float* xr1 = X + (size_t)(m + 16) * K;
  const float* wr  = W + (size_t)(nt * 16 + m) * K;

  v8f c0 = {}; v8f c1 = {};
  #pragma unroll 4
  for (int kb = 0; kb < K; kb += 4) {
    const int off = kb + 2 * hi;
    v2f a0 = *(const v2f*)(xr0 + off);
    v2f a1 = *(const v2f*)(xr1 + off);
    v2f bb = *(const v2f*)(wr + off);
    c0 = __builtin_amdgcn_wmma_f32_16x16x4_f32(false, a0, false, bb, (short)0, c0, false, false);
    c1 = __builtin_amdgcn_wmma_f32_16x16x4_f32(false, a1, false, bb, (short)0, c1, false, false);
  }

  const int col = nt * 16 + m;
  #pragma unroll
  for (int r = 0; r < 8; ++r) {
    Y[(size_t)(r + hi * 8) * N + col]      = c0[r];
    Y[(size_t)(r + hi * 8 + 16) * N + col] = c1[r];
  }
}

// ---------------------------------------------------------------------------
// RoPE on workspace Q and K (s == 1 -> single freqs row).
// ---------------------------------------------------------------------------
__global__ void rope_qk(float* __restrict__ xq, float* __restrict__ xk,
                        const float* __restrict__ freqs) {
  const int total = B_ * (NH_ + NKV_) * (HD_ / 2);
  int tid = blockIdx.x * blockDim.x + threadIdx.x;
  if (tid >= total) return;
  int i = tid & 63;
  int h = (tid >> 6) % (NH_ + NKV_);
  int b = tid / (64 * (NH_ + NKV_));
  float c  = freqs[i * 2 + 0];
  float sn = freqs[i * 2 + 1];
  float* p = (h < NH_) ? (xq + (size_t)(b * NH_ + h) * HD_)
                       : (xk + (size_t)(b * NKV_ + (h - NH_)) * HD_);
  float x0 = p[2 * i], x1 = p[2 * i + 1];
  p[2 * i]     = x0 * c - x1 * sn;
  p[2 * i + 1] = x0 * sn + x1 * c;
}

// ---------------------------------------------------------------------------
// Stage one 16-key K/V tile into LDS: one async b128 per row (coalesced 512B),
// row 4095 patched to the freshly projected k/v (cache inputs stay immutable).
// ---------------------------------------------------------------------------
__device__ __forceinline__ void stage_kv_tile(
    const float* kb0, const float* vb0, const float* knew, const float* vnew,
    int key0, int lane, float (*ktb)[132], float (*vtb)[136]) {
  const int c = lane * 4;
  #pragma unroll
  for (int r = 0; r < 16; ++r) {
    const int key = key0 + r;
    const float* ks = (key == L_ - 1) ? knew : (kb0 + (size_t)key * KVSTRIDE_);
    const float* vs = (key == L_ - 1) ? vnew : (vb0 + (size_t)key * KVSTRIDE_);
    async_b128(ks + c, &ktb[r][c]);
    async_b128(vs + c, &vtb[r][c]);
  }
}

// ---------------------------------------------------------------------------
// Flash-decode attention, split-K: one wave per (kv-head g, split sp, batch b).
// Async double-buffered K/V tiles in LDS; WMMA scores + online softmax +
// WMMA P*V; partial (O, m, l) written for the combine pass.
// ---------------------------------------------------------------------------
__global__ __launch_bounds__(32) void attn_decode(
    const float* __restrict__ xq, const float* __restrict__ xk,
    const float* __restrict__ xv, const float* __restrict__ cache_k,
    const float* __restrict__ cache_v, float* __restrict__ part_o,
    float* __restrict__ part_m, float* __restrict__ part_l) {
  const int g    = blockIdx.x;
  const int sp   = blockIdx.y;
  const int b    = blockIdx.z;
  const int lane = threadIdx.x;
  const int m    = lane & 15;
  const int hi   = lane >> 4;

  __shared__ float qs[16][130];     // banks: off + 2m  -> conflict-free
  __shared__ float pt[16][18];
  __shared__ float kt[2][16][132];  // banks: off + 4m  -> conflict-free
  __shared__ float vt[2][16][136];  // row pairs on disjoint bank ranges

  // Q: 4 valid heads, zero-padded to 16 rows (WMMA runs with full EXEC).
  for (int i = lane; i < 16 * HD_; i += 32) {
    int r = i >> 7, c = i & (HD_ - 1);
    qs[r][c] = (r < NREP_) ? xq[(size_t)(b * NH_ + g * NREP_ + r) * HD_ + c] : 0.0f;
  }
  __syncthreads();

  const float* kb0  = cache_k + (size_t)b * L_ * KVSTRIDE_ + (size_t)g * HD_;
  const float* vb0  = cache_v + (size_t)b * L_ * KVSTRIDE_ + (size_t)g * HD_;
  const float* knew = xk + (size_t)(b * NKV_ + g) * HD_;
  const float* vnew = xv + (size_t)(b * NKV_ + g) * HD_;

  v8f o[8] = {};
  float mrun[8], lrun[8];
  #pragma unroll
  for (int r = 0; r < 8; ++r) { mrun[r] = -1e30f; lrun[r] = 0.0f; }

  const float scale = 0.08838834764831845f;  // 1/sqrt(128)
  const int key_base = sp * (L_ / SPLIT_);

  stage_kv_tile(kb0, vb0, knew, vnew, key_base, lane, kt[0], vt[0]);

  for (int t = 0; t < TILES_; ++t) {
    const int key0 = key_base + t * 16;
    if (t + 1 < TILES_) {
      wait_dscnt0();  // prior reads of the buffer we overwrite have completed
      stage_kv_tile(kb0, vb0, knew, vnew, key0 + 16, lane,
                    kt[(t + 1) & 1], vt[(t + 1) & 1]);
      wait_async<32>();   // tile t's 32 async copies done (in-order completion)
    } else {
      wait_async<0>();
    }
    const float (*ktb)[132] = kt[t & 1];
    const float (*vtb)[136] = vt[t & 1];

    // S = Q(16x128) * K_tile^T(128x16), B operand streamed from LDS
    v8f s = {};
    #pragma unroll 8
    for (int d0 = 0; d0 < HD_; d0 += 4) {
      const int off = d0 + 2 * hi;
      v2f a  = *(const v2f*)&qs[m][off];
      v2f bv = *(const v2f*)&ktb[m][off];
      s = __builtin_amdgcn_wmma_f32_16x16x4_f32(false, a, false, bv, (short)0, s, false, false);
    }

    // Online softmax per row (row r spans one 16-lane half in VGPR r)
    float palpha[8];
    #pragma unroll
    for (int r = 0; r < 8; ++r) {
      float v = s[r] * scale;
      float tmax = v;
      tmax = fmaxf(tmax, __shfl_xor(tmax, 1, 32));
      tmax = fmaxf(tmax, __shfl_xor(tmax, 2, 32));
      tmax = fmaxf(tmax, __shfl_xor(tmax, 4, 32));
      tmax = fmaxf(tmax, __shfl_xor(tmax, 8, 32));
      float mnew = fmaxf(mrun[r], tmax);
      float p  = __expf(v - mnew);
      float ts = p;
      ts += __shfl_xor(ts, 1, 32);
      ts += __shfl_xor(ts, 2, 32);
      ts += __shfl_xor(ts, 4, 32);
      ts += __shfl_xor(ts, 8, 32);
      float al  = __expf(mrun[r] - mnew);
      lrun[r]   = lrun[r] * al + ts;
      mrun[r]   = mnew;
      palpha[r] = al;
      s[r] = p;
    }
    #pragma unroll
    for (int dt = 0; dt < 8; ++dt)
      #pragma unroll
      for (int r = 0; r < 8; ++r)
        o[dt][r] *= palpha[r];

    // C-layout -> LDS -> A-layout for the P*V WMMAs
    __syncthreads();
    #pragma unroll
    for (int r = 0; r < 8; ++r) pt[r + hi * 8][m] = s[r];
    __syncthreads();

    #pragma unroll
    for (int k0 = 0; k0 < 16; k0 += 4) {
      const int off = k0 + 2 * hi;
      v2f a = *(const v2f*)&pt[m][off];
      #pragma unroll
      for (int dt = 0; dt < 8; ++dt) {
        v2f bv; bv.x = vtb[off][dt * 16 + m]; bv.y = vtb[off + 1][dt * 16 + m];
        o[dt] = __builtin_amdgcn_wmma_f32_16x16x4_f32(false, a, false, bv, (short)0, o[dt], false, false);
      }
    }
  }

  // Store split partials (rows 0..3 are real heads: lanes 0..15, VGPRs 0..3)
  const size_t pbase = (size_t)((b * NKV_ + g) * SPLIT_ + sp) * NREP_;
  if (hi == 0) {
    #pragma unroll
    for (int r = 0; r < NREP_; ++r)
      #pragma unroll
      for (int dt = 0; dt < 8; ++dt)
        part_o[(pbase + r) * HD_ + dt * 16 + m] = o[dt][r];
  }
  if (lane == 0) {
    #pragma unroll
    for (int r = 0; r < NREP_; ++r) {
      part_m[pbase + r] = mrun[r];
      part_l[pbase + r] = lrun[r];
    }
  }
}

// ---------------------------------------------------------------------------
// Combine split-K partials: standard flash-decode merge.
// ---------------------------------------------------------------------------
__global__ __launch_bounds__(128) void attn_combine(
    const float* __restrict__ part_o, const float* __restrict__ part_m,
    const float* __restrict__ part_l, float* __restrict__ attn) {
  const int bh = blockIdx.x;            // b*NH + h
  const int b = bh / NH_, h = bh % NH_;
  const int g = h / NREP_, r = h % NREP_;
  const int d = threadIdx.x;            // 0..127
  const int base = (b * NKV_ + g) * SPLIT_;

  float ms[SPLIT_], ls[SPLIT_];
  float mg = -1e30f;
  #pragma unroll
  for (int i = 0; i < SPLIT_; ++i) {
    ms[i] = part_m[(base + i) * NREP_ + r];
    ls[i] = part_l[(base + i) * NREP_ + r];
    mg = fmaxf(mg, ms[i]);
  }
  float L = 0.0f, acc = 0.0f;
  #pragma unroll
  for (int i = 0; i < SPLIT_; ++i) {
    float w = __expf(ms[i] - mg);
    L   += ls[i] * w;
    acc += w * part_o[((size_t)(base + i) * NREP_ + r) * HD_ + d];
  }
  attn[(size_t)bh * HD_ + d] = acc / L;
}

// ---------------------------------------------------------------------------
extern "C" void kernel_launch(void* const* d_in, const int* in_sizes, int n_in,
                              void* d_out, int out_size, void* d_ws, size_t ws_size,
                              hipStream_t stream) {
  const float* x       = (const float*)d_in[0];
  const float* freqs   = (const float*)d_in[1];
  const float* cache_k = (const float*)d_in[2];
  const float* cache_v = (const float*)d_in[3];
  const float* wq      = (const float*)d_in[4];
  const float* wk      = (const float*)d_in[5];
  const float* wv      = (const float*)d_in[6];
  const float* wo      = (const float*)d_in[7];
  float* out = (float*)d_out;

  float* xq     = (float*)d_ws;                                   // 32*4096
  float* xk     = xq + (size_t)B_ * NH_ * HD_;                    // 32*1024
  float* xv     = xk + (size_t)B_ * NKV_ * HD_;                   // 32*1024
  float* part_o = xv + (size_t)B_ * NKV_ * HD_;                   // 32*8*4*4*128
  float* part_m = part_o + (size_t)B_ * NKV_ * SPLIT_ * NREP_ * HD_;
  float* part_l = part_m + (size_t)B_ * NKV_ * SPLIT_ * NREP_;
  float* attn   = part_l + (size_t)B_ * NKV_ * SPLIT_ * NREP_;    // 32*4096

  gemm_xwt<<<dim3((NH_ * HD_) / 16),  32, 0, stream>>>(x, wq, xq, DIM_, NH_ * HD_);
  gemm_xwt<<<dim3((NKV_ * HD_) / 16), 32, 0, stream>>>(x, wk, xk, DIM_, NKV_ * HD_);
  gemm_xwt<<<dim3((NKV_ * HD_) / 16), 32, 0, stream>>>(x, wv, xv, DIM_, NKV_ * HD_);

  const int pairs = B_ * (NH_ + NKV_) * (HD_ / 2);
  rope_qk<<<(pairs + 255) / 256, 256, 0, stream>>>(xq, xk, freqs);

  attn_decode<<<dim3(NKV_, SPLIT_, B_), 32, 0, stream>>>(
      xq, xk, xv, cache_k, cache_v, part_o, part_m, part_l);

  attn_combine<<<B_ * NH_, 128, 0, stream>>>(part_o, part_m, part_l, attn);

  gemm_xwt<<<dim3(DIM_ / 16), 32, 0, stream>>>(attn, wo, out, NH_ * HD_, DIM_);
}